// GCN_44263932953222
// MI455X (gfx1250) — compile-verified
//
#include <hip/hip_runtime.h>

// ---------------------------------------------------------------------------
// GCN forward for gfx1250 (MI455X). fp32 WMMA (V_WMMA_F32_16X16X4_F32) GEMMs
// with B staged in LDS, norm factored as dinv[src]*dinv[dst] -> pre/post
// scale, float4-vectorized atomicAdd scatter for the memory-bound aggregation.
// ---------------------------------------------------------------------------

typedef float v2f __attribute__((ext_vector_type(2)));
typedef float v8f __attribute__((ext_vector_type(8)));

#define TB 256

__global__ void k_zero4(float4* __restrict__ p, int n4) {
    int i = blockIdx.x * blockDim.x + threadIdx.x;
    if (i < n4) p[i] = make_float4(0.f, 0.f, 0.f, 0.f);
}

// deg[dst] += 1 per edge (self-loop folded in later as +1)
__global__ void k_deg(const int* __restrict__ dst, float* __restrict__ deg, int E) {
    int i = blockIdx.x * blockDim.x + threadIdx.x;
    if (i < E) atomicAdd(deg + dst[i], 1.0f);
}

// deg -> dinv = rsqrt(deg + 1)   (+1 = self loop; always > 0)
__global__ void k_dinv(float* __restrict__ deg, int N) {
    int i = blockIdx.x * blockDim.x + threadIdx.x;
    if (i < N) deg[i] = rsqrtf(deg[i] + 1.0f);
}

// Copy KxF row-major weight into zeroed Kloop x Fpad padded buffer
__global__ void k_copy_pad(const float* __restrict__ W, float* __restrict__ Wp,
                           int K, int F, int Fpad) {
    int i = blockIdx.x * blockDim.x + threadIdx.x;
    if (i >= K * F) return;
    int k = i / F, f = i - k * F;
    Wp[k * Fpad + f] = W[i];
}

// C[M x Fpad] = A[M x K] * Wp[Kloop x Fpad], one wave per 16x16 tile.
// fp32 WMMA, K-step 4. B tile staged in LDS (shared by all waves in block).
// Optional epilogue scale by dinv[row].
// Fragment layouts (CDNA5 ISA 7.12.2):
//   A 16x4 : lanes 0-15 row M=l hold K={k,k+1} in v[0],v[1]; lanes 16-31 K={k+2,k+3}
//   B 4x16 : v[0]: lanes0-15 row K=k, lanes16-31 row K=k+2; v[1]: K=k+1 / K=k+3
//   C/D    : VGPR r: lanes0-15 -> M=r, lanes16-31 -> M=r+8; N = lane&15
__global__ void k_gemm_wmma(const float* __restrict__ A, int lda,
                            const float* __restrict__ Wp, int ldw,
                            float* __restrict__ C, int ldc,
                            const float* __restrict__ dinv,
                            int Mtiles, int Ftiles, int Kloop) {
    extern __shared__ float sB[];                 // Kloop * ldw floats (<= 48.6 KB)
    const int nB = Kloop * ldw;
    for (int i = threadIdx.x; i < nB; i += blockDim.x) sB[i] = Wp[i];
    __syncthreads();

    const int lane = threadIdx.x & 31;
    const int wave = blockIdx.x * (blockDim.x >> 5) + (threadIdx.x >> 5);
    if (wave >= Mtiles * Ftiles) return;          // wave-uniform: EXEC stays all-1s

    const int mt = wave / Ftiles;
    const int ft = wave - mt * Ftiles;
    const int row0 = mt << 4;
    const int col0 = ft << 4;
    const int half = lane >> 4;                   // K sub-pair select
    const int l    = lane & 15;

    const float* __restrict__ arow = A + (size_t)(row0 + l) * lda + 2 * half;
    const float* bcol = sB + 2 * half * ldw + col0 + l;

    v8f acc = {};
    for (int k = 0; k < Kloop; k += 4) {
        v2f a = *(const v2f*)(arow + k);          // K = k+2*half, +1 (8B aligned)
        v2f b;
        b.x = bcol[k * ldw];                      // row K = k   + 2*half (LDS)
        b.y = bcol[(k + 1) * ldw];                // row K = k+1 + 2*half (LDS)
        acc = __builtin_amdgcn_wmma_f32_16x16x4_f32(
            /*neg_a=*/false, a, /*neg_b=*/false, b,
            /*c_mod=*/(short)0, acc, /*reuse_a=*/false, /*reuse_b=*/false);
    }

#pragma unroll
    for (int r = 0; r < 8; ++r) {
        int row = row0 + r + (half << 3);
        float v = acc[r];
        if (dinv) v *= dinv[row];
        C[(size_t)row * ldc + col0 + l] = v;
    }
}

// acc[dst, 4c..4c+3] += hs[src, 4c..4c+3] over E real edges + N self-loops.
// Covers the full padded width; padding columns are exact zeros in hs.
__global__ void k_scatter4(const float* __restrict__ hs, float* __restrict__ acc,
                           const int* __restrict__ src, const int* __restrict__ dst,
                           int E, int N, int C4, int ldh) {
    long long idx = blockIdx.x * (long long)blockDim.x + threadIdx.x;
    long long total = (long long)(E + N) * C4;
    if (idx >= total) return;
    int e = (int)(idx / C4);
    int c = (int)(idx - (long long)e * C4);
    int s, d;
    if (e < E) { s = src[e]; d = dst[e]; }
    else       { s = d = e - E; }
    const float4 v = *(const float4*)(hs + (size_t)s * ldh + 4 * c);  // b128 load
    float* p = acc + (size_t)d * ldh + 4 * c;
    atomicAdd(p + 0, v.x);
    atomicAdd(p + 1, v.y);
    atomicAdd(p + 2, v.z);
    atomicAdd(p + 3, v.w);
}

// h[node,f] = relu(dinv[node]*acc[node,f] + b[f])
__global__ void k_bias_relu(const float* __restrict__ acc, const float* __restrict__ b,
                            const float* __restrict__ dinv, float* __restrict__ h,
                            int N, int F, int ldh) {
    int i = blockIdx.x * blockDim.x + threadIdx.x;
    if (i >= N * F) return;
    int node = i / F, f = i - node * F;
    float v = fmaf(dinv[node], acc[(size_t)node * ldh + f], b[f]);
    h[(size_t)node * ldh + f] = fmaxf(v, 0.0f);
}

// order-preserving f32 -> u32 encoding for atomicMax
__device__ __forceinline__ unsigned enc_f32(float x) {
    unsigned b = __float_as_uint(x);
    return (b & 0x80000000u) ? ~b : (b | 0x80000000u);
}

__global__ void k_pool_max(const float* __restrict__ logits, int ldl,
                           const float* __restrict__ bo,
                           const int* __restrict__ batch,
                           unsigned* __restrict__ pooled, int N) {
    int i = blockIdx.x * blockDim.x + threadIdx.x;
    if (i >= N * 3) return;
    int node = i / 3, c = i - node * 3;
    float v = logits[(size_t)node * ldl + c] + bo[c];
    atomicMax(pooled + batch[node] * 3 + c, enc_f32(v));
}

__global__ void k_softmax(const unsigned* __restrict__ pooled, float* __restrict__ out) {
    int g = threadIdx.x;
    if (g >= 64) return;
    float v[3];
#pragma unroll
    for (int c = 0; c < 3; ++c) {
        unsigned key = pooled[g * 3 + c];
        unsigned bits = (key & 0x80000000u) ? (key & 0x7FFFFFFFu) : ~key;
        v[c] = __uint_as_float(bits);
    }
    float m = fmaxf(v[0], fmaxf(v[1], v[2]));
    float e0 = expf(v[0] - m), e1 = expf(v[1] - m), e2 = expf(v[2] - m);
    float inv = 1.0f / (e0 + e1 + e2);
    out[g * 3 + 0] = e0 * inv;
    out[g * 3 + 1] = e1 * inv;
    out[g * 3 + 2] = e2 * inv;
}

static inline unsigned nblk(long long n, int tb) { return (unsigned)((n + tb - 1) / tb); }

extern "C" void kernel_launch(void* const* d_in, const int* in_sizes, int n_in,
                              void* d_out, int out_size, void* d_ws, size_t ws_size,
                              hipStream_t stream) {
    const float* x     = (const float*)d_in[0];
    const int*   ei    = (const int*)d_in[1];
    const int*   batch = (const int*)d_in[2];
    const float* Wl[3] = {(const float*)d_in[3], (const float*)d_in[5], (const float*)d_in[7]};
    const float* bl[3] = {(const float*)d_in[4], (const float*)d_in[6], (const float*)d_in[8]};
    const float* Wo    = (const float*)d_in[9];
    const float* bo    = (const float*)d_in[10];

    const int N = in_sizes[2];        // 50000 (multiple of 16)
    const int E = in_sizes[1] / 2;    // 400000
    const int* src = ei;
    const int* dst = ei + E;

    // Workspace layout (floats)
    float* ws = (float*)d_ws;
    const size_t offD  = ((size_t)N + 63) & ~(size_t)63;   // dinv region, aligned
    const size_t HBUF  = (size_t)N * 160;                  // widest padded layer
    float* dinv  = ws;
    float* hGemm = ws + offD;
    float* hAcc  = hGemm + HBUF;
    float* hCur  = hAcc + HBUF;
    float* Wpad  = hCur + HBUF;                            // <= 152*64 / 76*160 floats
    unsigned* pooled = (unsigned*)(Wpad + 24320);          // 192 entries

    // Zero everything we use (multiple of 4 floats by construction)
    long long totalClear = (long long)offD + 3 * (long long)HBUF + 24320 + 192;
    k_zero4<<<nblk(totalClear / 4, TB), TB, 0, stream>>>((float4*)ws, (int)(totalClear / 4));

    // Degrees -> dinv
    k_deg<<<nblk(E, TB), TB, 0, stream>>>(dst, dinv, E);
    k_dinv<<<nblk(N, TB), TB, 0, stream>>>(dinv, N);

    const int Kdim[3] = {36, 75, 150};
    const int Fdim[3] = {75, 150, 50};
    const int Fpad[3] = {80, 160, 64};
    const int Mtiles  = N / 16;

    const float* Ain = x;
    int lda = 36;
    for (int li = 0; li < 3; ++li) {
        const int K = Kdim[li], F = Fdim[li], Fp = Fpad[li];
        const int Kloop = (K + 3) & ~3;
        // stage padded weights (zero padding rows/cols annihilate activation padding)
        k_zero4<<<nblk((long long)Kloop * Fp / 4, TB), TB, 0, stream>>>(
            (float4*)Wpad, Kloop * Fp / 4);
        k_copy_pad<<<nblk((long long)K * F, TB), TB, 0, stream>>>(Wl[li], Wpad, K, F, Fp);
        // hGemm = (Ain @ W) * dinv[row]     (WMMA, B tile in LDS)
        const int Ftiles = Fp / 16;
        long long waves = (long long)Mtiles * Ftiles;
        size_t smem = (size_t)Kloop * Fp * sizeof(float);  // <= 48,640 B
        k_gemm_wmma<<<nblk(waves * 32, TB), TB, smem, stream>>>(
            Ain, lda, Wpad, Fp, hGemm, Fp, dinv, Mtiles, Ftiles, Kloop);
        // acc = 0 ; acc[dst] += hGemm[src] over edges + self loops (float4 chunks)
        k_zero4<<<nblk((long long)N * Fp / 4, TB), TB, 0, stream>>>(
            (float4*)hAcc, N * Fp / 4);
        k_scatter4<<<nblk((long long)(E + N) * (Fp / 4), TB), TB, 0, stream>>>(
            hGemm, hAcc, src, dst, E, N, Fp / 4, Fp);
        // hCur = relu(dinv[dst]*acc + b)
        k_bias_relu<<<nblk((long long)N * F, TB), TB, 0, stream>>>(
            hAcc, bl[li], dinv, hCur, N, F, Fp);
        Ain = hCur;
        lda = Fp;
    }

    // logits = h3 @ Wo  (bias folded into pooling), Fpad=16, Kloop=52
    k_zero4<<<nblk(52 * 16 / 4, TB), TB, 0, stream>>>((float4*)Wpad, 52 * 16 / 4);
    k_copy_pad<<<nblk(50 * 3, TB), TB, 0, stream>>>(Wo, Wpad, 50, 3, 16);
    k_gemm_wmma<<<nblk((long long)Mtiles * 32, TB), TB, 52 * 16 * sizeof(float), stream>>>(
        hCur, 64, Wpad, 16, hGemm, 16, nullptr, Mtiles, 1, 52);

    // segment_max over batch (+bo), then softmax -> d_out [64 x 3]
    k_pool_max<<<nblk((long long)N * 3, TB), TB, 0, stream>>>(hGemm, 16, bo, batch, pooled, N);
    k_softmax<<<1, 64, 0, stream>>>(pooled, (float*)d_out);

    (void)n_in; (void)out_size; (void)ws_size;
}